// HypMultiProxyAnchor_24988119728223
// MI455X (gfx1250) — compile-verified
//
#include <hip/hip_runtime.h>
#include <hip/hip_bf16.h>
#include <math.h>

typedef __attribute__((ext_vector_type(2))) float v2f;
typedef __attribute__((ext_vector_type(8))) float v8f;

#define CVAL   0.1f
#define SQRTC  0.31622776601683794f
#define CLIPR  2.3f
#define NG     60
#define NGP    64      // genus padded to 64 for 16x16 tiling
#define NS     4096
#define NB     1024
#define EMBD   64

// ---------- helpers ----------
__device__ __forceinline__ float waveSum32(float v) {
#pragma unroll
    for (int m = 16; m > 0; m >>= 1) v += __shfl_xor(v, m, 32);
    return v;
}

__device__ __forceinline__ float softplusf(float x) {
    // log(1+exp(x)) stable
    return fmaxf(x, 0.f) + log1pf(expf(-fabsf(x)));
}

// Poincare distance from dot(X,y), ||x||^2, ||y||^2 (a = -x, dot(a,y) = -dotXY)
__device__ __forceinline__ float hyp_dist(float dotXY, float a2, float y2) {
    float dotp = -dotXY;
    float ca   = 1.f + 2.f * CVAL * dotp + CVAL * y2;
    float cy   = 1.f - CVAL * a2;
    float num2 = ca * ca * a2 + 2.f * ca * cy * dotp + cy * cy * y2;
    float den  = fmaxf(fabsf(1.f + 2.f * CVAL * dotp + CVAL * CVAL * a2 * y2), 1e-15f);
    float mn   = sqrtf(fmaxf(num2, 0.f)) / den;
    float arg  = fminf(SQRTC * mn, 1.f - 1e-5f);       // >= 0 by construction
    return (2.f / SQRTC) * atanhf(arg);
}

// ---------- kernels ----------
__global__ void zero_f32(float* p, int n) {
    int i = blockIdx.x * blockDim.x + threadIdx.x;
    if (i < n) p[i] = 0.f;
}

// HypProjector: one wave per row of 64 floats (2 elems/lane), plus ||y||^2.
__global__ __launch_bounds__(256) void project_rows(const float* __restrict__ in,
                                                    float* __restrict__ out,
                                                    float* __restrict__ y2out, int nrows) {
    int w = (blockIdx.x * 256 + threadIdx.x) >> 5;
    int lane = threadIdx.x & 31;
    if (w >= nrows) return;                 // wave-uniform
    const float* xr = in + w * EMBD;
    float x0 = xr[lane], x1 = xr[lane + 32];
    float n = fmaxf(sqrtf(waveSum32(x0 * x0 + x1 * x1)), 1e-5f);
    float s = fminf(1.f, CLIPR / n);
    x0 *= s; x1 *= s;
    float n2 = fmaxf(sqrtf(waveSum32(x0 * x0 + x1 * x1)), 1e-5f);
    float t  = tanhf(SQRTC * n2) / (SQRTC * n2);
    float y0 = t * x0, y1 = t * x1;
    float yn = fmaxf(sqrtf(waveSum32(y0 * y0 + y1 * y1)), 1e-5f);
    float maxn = (1.f - 1e-3f) / SQRTC;
    if (yn > maxn) { float r = maxn / yn; y0 *= r; y1 *= r; }
    out[w * EMBD + lane] = y0;
    out[w * EMBD + lane + 32] = y1;
    float y2 = waveSum32(y0 * y0 + y1 * y1);
    if (lane == 0) y2out[w] = y2;
}

// ||row||^2 for X (a2 in dist_matrix)
__global__ __launch_bounds__(256) void row_norm2(const float* __restrict__ in,
                                                 float* __restrict__ out, int nrows) {
    int w = (blockIdx.x * 256 + threadIdx.x) >> 5;
    int lane = threadIdx.x & 31;
    if (w >= nrows) return;
    const float* xr = in + w * EMBD;
    float v = waveSum32(xr[lane] * xr[lane] + xr[lane + 32] * xr[lane + 32]);
    if (lane == 0) out[w] = v;
}

__global__ void scatter_present(const int* __restrict__ Ts, float* __restrict__ present) {
    int b = blockIdx.x * blockDim.x + threadIdx.x;
    if (b < NB) present[Ts[b]] = 1.f;       // idempotent store, race-free semantics
}

__global__ void scatter_validg(const int* __restrict__ s2g, const float* __restrict__ present,
                               float* __restrict__ validg) {
    int s = blockIdx.x * blockDim.x + threadIdx.x;
    if (s < NS && present[s] != 0.f) validg[s2g[s]] = 1.f;
}

// Species loss tiles: one 16x16 tile per wave, K=64 via 16x V_WMMA_F32_16X16X4_F32,
// fused Poincare distance + softplus epilogue, block partials out.
__global__ __launch_bounds__(256) void dist_species(const float* __restrict__ X,
                                                    const float* __restrict__ hs,
                                                    const float* __restrict__ a2x,
                                                    const float* __restrict__ y2s,
                                                    const int* __restrict__ Ts,
                                                    float* __restrict__ part) {
    __shared__ float red[3][256];
    const int lane = threadIdx.x & 31;
    const int wv   = threadIdx.x >> 5;
    const int tile = blockIdx.x * 8 + wv;   // 64 x 256 tiles, exact
    const int tr = tile >> 8;
    const int tc = tile & 255;
    const int lo = lane & 15, hi = lane >> 4;
    const float* Ap = X  + (tr * 16 + lo) * EMBD + 2 * hi;   // A[M=lo][K=2*hi+{0,1}]
    const float* Bp = hs + (tc * 16 + lo) * EMBD + 2 * hi;   // B[K][N=lo] = hs[N][K]
    v8f acc = {};
#pragma unroll
    for (int kk = 0; kk < 16; ++kk) {
        v2f a = *(const v2f*)(Ap + kk * 4);
        v2f b = *(const v2f*)(Bp + kk * 4);
        acc = __builtin_amdgcn_wmma_f32_16x16x4_f32(false, a, false, b,
                                                    (short)0, acc, false, false);
    }
    const int col = tc * 16 + lo;
    const float y2 = y2s[col];
    float nAll = 0.f, pPos = 0.f, nPos = 0.f;
#pragma unroll
    for (int i = 0; i < 8; ++i) {
        int row = tr * 16 + i + 8 * hi;
        float d  = hyp_dist(acc[i], a2x[row], y2);
        float sn = softplusf(-d);
        nAll += sn;
        if (Ts[row] == col) { pPos += softplusf(d); nPos += sn; }
    }
    red[0][threadIdx.x] = nAll; red[1][threadIdx.x] = pPos; red[2][threadIdx.x] = nPos;
    __syncthreads();
    for (int s = 128; s > 0; s >>= 1) {
        if (threadIdx.x < s) {
            red[0][threadIdx.x] += red[0][threadIdx.x + s];
            red[1][threadIdx.x] += red[1][threadIdx.x + s];
            red[2][threadIdx.x] += red[2][threadIdx.x + s];
        }
        __syncthreads();
    }
    if (threadIdx.x == 0) {
        part[blockIdx.x * 3 + 0] = red[0][0];
        part[blockIdx.x * 3 + 1] = red[1][0];
        part[blockIdx.x * 3 + 2] = red[2][0];
    }
}

// Genus loss tiles: hyp_s [4096,64] x hyp_g padded [64,64]; cols >= 60 masked.
__global__ __launch_bounds__(256) void dist_genus(const float* __restrict__ hs,
                                                  const float* __restrict__ hg,
                                                  const float* __restrict__ y2s,
                                                  const float* __restrict__ y2g,
                                                  const float* __restrict__ present,
                                                  const int* __restrict__ s2g,
                                                  float* __restrict__ part) {
    __shared__ float red[3][256];
    const int lane = threadIdx.x & 31;
    const int wv   = threadIdx.x >> 5;
    const int tile = blockIdx.x * 8 + wv;   // 256 x 4 tiles, exact
    const int tr = tile >> 2;
    const int tc = tile & 3;
    const int lo = lane & 15, hi = lane >> 4;
    const float* Ap = hs + (tr * 16 + lo) * EMBD + 2 * hi;
    const float* Bp = hg + (tc * 16 + lo) * EMBD + 2 * hi;
    v8f acc = {};
#pragma unroll
    for (int kk = 0; kk < 16; ++kk) {
        v2f a = *(const v2f*)(Ap + kk * 4);
        v2f b = *(const v2f*)(Bp + kk * 4);
        acc = __builtin_amdgcn_wmma_f32_16x16x4_f32(false, a, false, b,
                                                    (short)0, acc, false, false);
    }
    const int col = tc * 16 + lo;
    const bool colOK = (col < NG);
    const float y2 = y2g[col];
    float nAll = 0.f, pPos = 0.f, nPos = 0.f;
#pragma unroll
    for (int i = 0; i < 8; ++i) {
        int row = tr * 16 + i + 8 * hi;
        if (colOK && present[row] != 0.f) {
            float d  = hyp_dist(acc[i], y2s[row], y2);
            float sn = softplusf(-d);
            nAll += sn;
            if (s2g[row] == col) { pPos += softplusf(d); nPos += sn; }
        }
    }
    red[0][threadIdx.x] = nAll; red[1][threadIdx.x] = pPos; red[2][threadIdx.x] = nPos;
    __syncthreads();
    for (int s = 128; s > 0; s >>= 1) {
        if (threadIdx.x < s) {
            red[0][threadIdx.x] += red[0][threadIdx.x + s];
            red[1][threadIdx.x] += red[1][threadIdx.x + s];
            red[2][threadIdx.x] += red[2][threadIdx.x + s];
        }
        __syncthreads();
    }
    if (threadIdx.x == 0) {
        part[blockIdx.x * 3 + 0] = red[0][0];
        part[blockIdx.x * 3 + 1] = red[1][0];
        part[blockIdx.x * 3 + 2] = red[2][0];
    }
}

// Deterministic final reduction: fixed-order strided sums + LDS tree.
__global__ __launch_bounds__(256) void finalize(const float* __restrict__ partS,
                                                const float* __restrict__ partG,
                                                const float* __restrict__ present,
                                                const float* __restrict__ validg,
                                                float* __restrict__ out) {
    __shared__ float red[256];
    float sums[8];
    for (int ch = 0; ch < 8; ++ch) {
        float v = 0.f;
        if (ch < 3)       { for (int i = threadIdx.x; i < 2048; i += 256) v += partS[i * 3 + ch]; }
        else if (ch < 6)  { for (int i = threadIdx.x; i < 128;  i += 256) v += partG[i * 3 + (ch - 3)]; }
        else if (ch == 6) { for (int i = threadIdx.x; i < NS;   i += 256) v += present[i]; }
        else              { for (int i = threadIdx.x; i < NG;   i += 256) v += validg[i]; }
        red[threadIdx.x] = v; __syncthreads();
        for (int s = 128; s > 0; s >>= 1) {
            if (threadIdx.x < s) red[threadIdx.x] += red[threadIdx.x + s];
            __syncthreads();
        }
        sums[ch] = red[0]; __syncthreads();
    }
    if (threadIdx.x == 0) {
        float nvs = fmaxf(sums[6], 1.f);
        float nvg = fmaxf(sums[7], 1.f);
        float loss = sums[1] / nvs + (sums[0] - sums[2]) / (float)NS
                   + sums[4] / nvg + (sums[3] - sums[5]) / (float)NG;
        out[0] = loss;
    }
}

extern "C" void kernel_launch(void* const* d_in, const int* in_sizes, int n_in,
                              void* d_out, int out_size, void* d_ws, size_t ws_size,
                              hipStream_t stream) {
    const float* X  = (const float*)d_in[0];
    const int*  Ts  = (const int*)d_in[1];
    // d_in[2] = T_genus (unused by the reference loss)
    const int*  s2g = (const int*)d_in[3];
    const float* gp = (const float*)d_in[4];
    const float* sp = (const float*)d_in[5];

    float* ws      = (float*)d_ws;
    float* hs      = ws;                    // 4096*64
    float* y2s     = hs + NS * EMBD;        // 4096
    float* hg      = y2s + NS;              // 64*64 (padded)
    float* y2g     = hg + NGP * EMBD;       // 64
    float* a2x     = y2g + NGP;             // 1024
    float* present = a2x + NB;              // 4096
    float* validg  = present + NS;          // 64
    float* partS   = validg + NGP;          // 2048*3
    float* partG   = partS + 2048 * 3;      // 128*3

    // init: zero padded genus block (+norms) and masks
    zero_f32<<<(NGP * EMBD + NGP + 255) / 256, 256, 0, stream>>>(hg, NGP * EMBD + NGP);
    zero_f32<<<(NS + NGP + 255) / 256, 256, 0, stream>>>(present, NS + NGP);

    project_rows<<<NS / 8, 256, 0, stream>>>(sp, hs, y2s, NS);       // 512 blocks
    project_rows<<<(NG + 7) / 8, 256, 0, stream>>>(gp, hg, y2g, NG); // 8 blocks
    row_norm2<<<NB / 8, 256, 0, stream>>>(X, a2x, NB);               // 128 blocks

    scatter_present<<<NB / 256, 256, 0, stream>>>(Ts, present);
    scatter_validg<<<NS / 256, 256, 0, stream>>>(s2g, present, validg);

    dist_species<<<2048, 256, 0, stream>>>(X, hs, a2x, y2s, Ts, partS);
    dist_genus<<<128, 256, 0, stream>>>(hs, hg, y2s, y2g, present, s2g, partG);

    finalize<<<1, 256, 0, stream>>>(partS, partG, present, validg, (float*)d_out);
}